// _MeshConv_78546361909484
// MI455X (gfx1250) — compile-verified
//
#include <hip/hip_runtime.h>

#define NVERT 163842
#define NFACE 327680
#define BB    2
#define CINC  16
#define COUTC 16
#define TILE_V 64
#define NTILES ((NVERT + TILE_V - 1) / TILE_V)   // 2561

typedef __attribute__((ext_vector_type(2))) float v2f;
typedef __attribute__((ext_vector_type(8))) float v8f;

// ---------------------------------------------------------------------------
// Kernel 1: per-face gradient + EW/NS reduction.
// grad_face[b,c,k,f] = sum_j G_vals[3*(k*NF+f)+j] * x[b,c, G_cols[3f+j]]
// gfe[b,c,f] = dot(grad_face[b,c,:,f], EW[f,:]);  gfn likewise with NS.
// One thread per face; loop the 32 (b,c) slices so the 9 G_vals, 3 cols and
// 6 EW/NS floats are loaded once per face instead of 32 times.
// ---------------------------------------------------------------------------
__global__ void face_grad_kernel(const float* __restrict__ x,
                                 const float* __restrict__ G_vals,
                                 const int*   __restrict__ G_cols,
                                 const float* __restrict__ NS,
                                 const float* __restrict__ EW,
                                 float* __restrict__ gfe,
                                 float* __restrict__ gfn) {
  int f = blockIdx.x * blockDim.x + threadIdx.x;
  if (f >= NFACE) return;

  const int v0 = G_cols[3 * f + 0];
  const int v1 = G_cols[3 * f + 1];
  const int v2 = G_cols[3 * f + 2];

  const float g00 = G_vals[3 * f + 0];
  const float g01 = G_vals[3 * f + 1];
  const float g02 = G_vals[3 * f + 2];
  const float g10 = G_vals[3 * (NFACE + f) + 0];
  const float g11 = G_vals[3 * (NFACE + f) + 1];
  const float g12 = G_vals[3 * (NFACE + f) + 2];
  const float g20 = G_vals[3 * (2 * NFACE + f) + 0];
  const float g21 = G_vals[3 * (2 * NFACE + f) + 1];
  const float g22 = G_vals[3 * (2 * NFACE + f) + 2];

  const float e0 = EW[3 * f + 0], e1 = EW[3 * f + 1], e2 = EW[3 * f + 2];
  const float s0 = NS[3 * f + 0], s1 = NS[3 * f + 1], s2 = NS[3 * f + 2];

#pragma unroll 4
  for (int bc = 0; bc < BB * CINC; ++bc) {
    const float* xs = x + bc * NVERT;
    const float x0 = xs[v0], x1 = xs[v1], x2 = xs[v2];
    const float k0 = g00 * x0 + g01 * x1 + g02 * x2;
    const float k1 = g10 * x0 + g11 * x1 + g12 * x2;
    const float k2 = g20 * x0 + g21 * x1 + g22 * x2;
    gfe[bc * NFACE + f] = k0 * e0 + k1 * e1 + k2 * e2;
    gfn[bc * NFACE + f] = k0 * s0 + k1 * s1 + k2 * s2;
  }
}

// ---------------------------------------------------------------------------
// Kernel 2: fused vertex gather + dense contraction via V_WMMA_F32_16X16X4_F32.
// Block = 128 threads (4 waves) handles one batch b and 64 vertices.
// Phase 1: compute feat[c][v][k] (k: identity, laplacian, grad_ew, grad_ns)
//          into LDS (16 KB).
// Phase 2: each wave owns a 16-vertex subtile and chains 16 f32 WMMAs
//          (one per input channel), accumulator seeded with bias.
//   A (16x4, M=o, K=k): lane<16 holds K=0,1; lane>=16 holds K=2,3; M=lane&15.
//   B (4x16, K=k, N=v): same half-wave K split, N=lane&15.
//   C/D (16x16): VGPR g -> M = g + (lane<16 ? 0 : 8), N = lane&15.
// ---------------------------------------------------------------------------
__global__ void vert_wmma_kernel(const float* __restrict__ x,
                                 const float* __restrict__ L_vals,
                                 const int*   __restrict__ L_cols,
                                 const float* __restrict__ F2V_vals,
                                 const int*   __restrict__ F2V_cols,
                                 const float* __restrict__ gfe,
                                 const float* __restrict__ gfn,
                                 const float* __restrict__ coeffs,
                                 const float* __restrict__ bias,
                                 float* __restrict__ out) {
  __shared__ float feat[CINC * TILE_V * 4];   // [c][v][k], 16 KB

  const int tid   = threadIdx.x;
  const int b     = blockIdx.x / NTILES;
  const int vbase = (blockIdx.x % NTILES) * TILE_V;

  // ---- Phase 1: gather 4 features per (c, v) into LDS -------------------
#pragma unroll
  for (int it = 0; it < (CINC * TILE_V) / 128; ++it) {
    const int t  = it * 128 + tid;
    const int c  = t >> 6;          // / TILE_V
    const int vl = t & (TILE_V - 1);
    int v = vbase + vl;
    const int vc = (v < NVERT) ? v : (NVERT - 1);   // clamp: tail columns are
                                                    // garbage but never stored
    const int bc = b * CINC + c;
    const float* xs = x + bc * NVERT;

    const float iden = xs[vc];

    float lap = 0.0f;
#pragma unroll
    for (int j = 0; j < 7; ++j)
      lap += L_vals[7 * vc + j] * xs[L_cols[7 * vc + j]];

    const float* ge = gfe + bc * NFACE;
    const float* gn = gfn + bc * NFACE;
    float ve = 0.0f, vn = 0.0f;
#pragma unroll
    for (int j = 0; j < 6; ++j) {
      const float w  = F2V_vals[6 * vc + j];
      const int   fc = F2V_cols[6 * vc + j];
      ve += w * ge[fc];
      vn += w * gn[fc];
    }

    float* dst = feat + (c * TILE_V + vl) * 4;
    dst[0] = iden;
    dst[1] = lap;
    dst[2] = ve;
    dst[3] = vn;
  }
  __syncthreads();

  // ---- Phase 2: 16 chained V_WMMA_F32_16X16X4_F32 per 16-vertex subtile --
  const int wave  = tid >> 5;
  const int lane  = tid & 31;
  const int row   = lane & 15;
  const int koff  = (lane < 16) ? 0 : 2;
  const int obase = (lane < 16) ? 0 : 8;
  const int vsub  = wave * 16;

  // Preload all channel slices of coeffs (16 x v2f = 32 VGPRs).
  v2f A[CINC];
#pragma unroll
  for (int c = 0; c < CINC; ++c) {
    const float* ap = coeffs + (row * CINC + c) * 4 + koff;  // coeffs[o][c][k]
    A[c].x = ap[0];
    A[c].y = ap[1];
  }

  v8f acc;
#pragma unroll
  for (int g = 0; g < 8; ++g) acc[g] = bias[obase + g];

#pragma unroll
  for (int c = 0; c < CINC; ++c) {
    const float* bp = feat + (c * TILE_V + vsub + row) * 4 + koff;
    v2f Bv;
    Bv.x = bp[0];
    Bv.y = bp[1];
    acc = __builtin_amdgcn_wmma_f32_16x16x4_f32(
        /*neg_a=*/false, A[c], /*neg_b=*/false, Bv,
        /*c_mod=*/(short)0, acc, /*reuse_a=*/false, /*reuse_b=*/false);
  }

  const int n = vbase + vsub + row;
  if (n < NVERT) {
    float* op = out + (b * COUTC + obase) * NVERT + n;
#pragma unroll
    for (int g = 0; g < 8; ++g) op[g * NVERT] = acc[g];
  }
}

// ---------------------------------------------------------------------------
// Launch. Input order (setup_inputs dict):
//  0 x, 1 G_vals, 2 L_vals, 3 F2V_vals, 4 NS, 5 EW, 6 coeffs, 7 bias,
//  8 G_rows, 9 G_cols, 10 L_rows, 11 L_cols, 12 F2V_rows, 13 F2V_cols
// (rows arrays are structured repeats; not needed.)
// ---------------------------------------------------------------------------
extern "C" void kernel_launch(void* const* d_in, const int* in_sizes, int n_in,
                              void* d_out, int out_size, void* d_ws, size_t ws_size,
                              hipStream_t stream) {
  (void)in_sizes; (void)n_in; (void)out_size; (void)ws_size;

  const float* x        = (const float*)d_in[0];
  const float* G_vals   = (const float*)d_in[1];
  const float* L_vals   = (const float*)d_in[2];
  const float* F2V_vals = (const float*)d_in[3];
  const float* NS       = (const float*)d_in[4];
  const float* EW       = (const float*)d_in[5];
  const float* coeffs   = (const float*)d_in[6];
  const float* bias     = (const float*)d_in[7];
  const int*   G_cols   = (const int*)d_in[9];
  const int*   L_cols   = (const int*)d_in[11];
  const int*   F2V_cols = (const int*)d_in[13];
  float* out = (float*)d_out;

  // Workspace: gfe [B*C*NF] floats, then gfn [B*C*NF] floats (~84 MB total).
  float* gfe = (float*)d_ws;
  float* gfn = gfe + (size_t)BB * CINC * NFACE;

  face_grad_kernel<<<(NFACE + 255) / 256, 256, 0, stream>>>(
      x, G_vals, G_cols, NS, EW, gfe, gfn);

  vert_wmma_kernel<<<BB * NTILES, 128, 0, stream>>>(
      x, L_vals, L_cols, F2V_vals, F2V_cols, gfe, gfn, coeffs, bias, out);
}